// ButterworthDecomposition_17815524344028
// MI455X (gfx1250) — compile-verified
//
#include <hip/hip_runtime.h>

// ---------------- problem constants ----------------
#define PADL   27          // 3*(2*4+1)
#define TLEN   8192
#define TEXT   8246        // TLEN + 2*PADL
#define NCH    2048        // 32*64 rows
#define KB     516         // ceil(TEXT/16)
#define Y1STR  8256        // padded row stride for intermediate (16-aligned)
#define QSTR   (KB * 8)    // per-chain q/z stride (floats)

// per-band constant block layout (floats)
#define CONST_F 1024
#define OFF_T   0          // T[16][16] row-major
#define OFF_P   256        // P[16][8]  row-major
#define OFF_A   384        // A16[8][8] row-major
#define OFF_Q   448        // Qc[16][8]: Qc[j][i] = state resp S_{16-j}[i]
#define OFF_ZI  576        // zi[8]
#define DATA_OFF 2048      // floats reserved for both bands' constants

typedef __attribute__((ext_vector_type(2))) float v2f;
typedef __attribute__((ext_vector_type(4))) float v4f;
typedef __attribute__((ext_vector_type(8))) float v8f;

// exact reference biquad-cascade step (transposed direct form II)
__device__ __forceinline__ float sos_step(const float* b, const float* a, float* z, float x) {
  float y = x;
#pragma unroll
  for (int s = 0; s < 4; ++s) {
    float out = b[3*s+0] * y + z[2*s+0];
    float z0  = b[3*s+1] * y - a[2*s+0] * out + z[2*s+1];
    float z1  = b[3*s+2] * y - a[2*s+1] * out;
    z[2*s+0] = z0; z[2*s+1] = z1; y = out;
  }
  return y;
}

// Build T, P, A16, Q, zi for each band from the SOS coefficients.
__global__ void setup_kernel(float* __restrict__ ws,
                             const float* __restrict__ sos_low,
                             const float* __restrict__ sos_high) {
  int band = threadIdx.x;
  if (band >= 2) return;
  const float* sos = band ? sos_high : sos_low;
  float* cst = ws + band * CONST_F;

  float b[12], a[8];
#pragma unroll
  for (int s = 0; s < 4; ++s) {
    b[3*s+0] = sos[6*s+0]; b[3*s+1] = sos[6*s+1]; b[3*s+2] = sos[6*s+2];
    a[2*s+0] = sos[6*s+4]; a[2*s+1] = sos[6*s+5];
  }

  // sosfilt_zi (steady-state initial conditions, cumulative gain scaling)
  float scale = 1.0f;
#pragma unroll
  for (int s = 0; s < 4; ++s) {
    float b0 = b[3*s], b1 = b[3*s+1], b2 = b[3*s+2], a1 = a[2*s], a2 = a[2*s+1];
    float B0 = b1 - a1 * b0, B1 = b2 - a2 * b0;
    float det = 1.0f + a1 + a2;
    cst[OFF_ZI + 2*s+0] = ((B0 + B1) / det) * scale;
    cst[OFF_ZI + 2*s+1] = (((1.0f + a1) * B1 - a2 * B0) / det) * scale;
    scale *= (b0 + b1 + b2) / det;
  }

  // impulse run: h[0..15]; state snapshots S_k -> Q columns (Q[:,j] = S_{16-j})
  float z[8], h[16];
#pragma unroll
  for (int i = 0; i < 8; ++i) z[i] = 0.0f;
  for (int t = 0; t < 16; ++t) {
    h[t] = sos_step(b, a, z, (t == 0) ? 1.0f : 0.0f);
    int j = 16 - (t + 1);
#pragma unroll
    for (int i = 0; i < 8; ++i) cst[OFF_Q + j*8 + i] = z[i];
  }
  for (int i = 0; i < 16; ++i)
    for (int j = 0; j < 16; ++j)
      cst[OFF_T + i*16 + j] = (i >= j) ? h[i - j] : 0.0f;

  // unit-state runs: P columns (state->output), A16 columns (state->state)
  for (int e = 0; e < 8; ++e) {
#pragma unroll
    for (int i = 0; i < 8; ++i) z[i] = 0.0f;
    z[e] = 1.0f;
    for (int t = 0; t < 16; ++t)
      cst[OFF_P + t*8 + e] = sos_step(b, a, z, 0.0f);
#pragma unroll
    for (int i = 0; i < 8; ++i) cst[OFF_A + i*8 + e] = z[i];
  }
}

// odd-reflection-extended sample t of row x[chain][*]; t >= TEXT -> 0  (edge path)
__device__ __forceinline__ float load_ext(const float* __restrict__ row, int t) {
  if (t >= TEXT) return 0.0f;
  if (t < PADL)        return 2.0f * row[0]        - row[PADL - t];
  if (t < PADL + TLEN) return row[t - PADL];
  int i = t - (PADL + TLEN);
  return 2.0f * row[TLEN - 1] - row[TLEN - 2 - i];
}

// q_k = Q * x_block  (zero-state end-of-block state contribution)
template <int PASS>
__global__ void q_kernel(const float* __restrict__ ws, const float* __restrict__ x,
                         const float* __restrict__ y1, float* __restrict__ qbuf,
                         int band) {
  int idx = blockIdx.x * blockDim.x + threadIdx.x;
  if (idx >= NCH * KB) return;
  int chain = idx / KB, k = idx % KB, t0 = k * 16;
  const float* qc = ws + band * CONST_F + OFF_Q;

  float xv[16];
  if (PASS == 0) {
    const float* row = x + (size_t)chain * TLEN;
    if (t0 >= PADL && t0 + 16 <= PADL + TLEN) {      // interior fast path
      const float* p = row + (t0 - PADL);
#pragma unroll
      for (int j = 0; j < 16; ++j) xv[j] = p[j];
    } else {
#pragma unroll
      for (int j = 0; j < 16; ++j) xv[j] = load_ext(row, t0 + j);
    }
  } else {
    const float* row = y1 + (size_t)chain * Y1STR;
    if (t0 + 16 <= TEXT) {                           // reversed fast path
      int u0 = TEXT - 1 - t0;
#pragma unroll
      for (int j = 0; j < 16; ++j) xv[j] = row[u0 - j];
    } else {
#pragma unroll
      for (int j = 0; j < 16; ++j) xv[j] = (t0 + j < TEXT) ? row[TEXT - 1 - t0 - j] : 0.0f;
    }
  }

  float q[8] = {0,0,0,0,0,0,0,0};
#pragma unroll
  for (int j = 0; j < 16; ++j) {
#pragma unroll
    for (int i = 0; i < 8; ++i) q[i] = fmaf(qc[j*8 + i], xv[j], q[i]);
  }
  float* qp = qbuf + (size_t)chain * QSTR + k * 8;   // 16B-aligned
  *(v4f*)(qp)     = (v4f){q[0], q[1], q[2], q[3]};
  *(v4f*)(qp + 4) = (v4f){q[4], q[5], q[6], q[7]};
}

// sequential state scan over blocks (parallel over 2048 chains), in-place q->z
__global__ void scan_kernel(const float* __restrict__ ws, const float* __restrict__ x,
                            const float* __restrict__ y1, float* __restrict__ qbuf,
                            int band, int pass) {
  int chain = blockIdx.x * blockDim.x + threadIdx.x;
  if (chain >= NCH) return;
  const float* cst = ws + band * CONST_F;
  float x0 = (pass == 0)
      ? (2.0f * x[(size_t)chain * TLEN] - x[(size_t)chain * TLEN + PADL])
      : y1[(size_t)chain * Y1STR + (TEXT - 1)];
  float z[8], A[64];
#pragma unroll
  for (int i = 0; i < 8; ++i) z[i] = x0 * cst[OFF_ZI + i];
#pragma unroll
  for (int i = 0; i < 16; ++i)
    *(v4f*)(A + 4*i) = *(const v4f*)(cst + OFF_A + 4*i);      // 16B-aligned
  float* qp = qbuf + (size_t)chain * QSTR;
  for (int k = 0; k < KB; ++k) {
    float q8[8], nz[8];
    v4f qa = *(const v4f*)(qp + k*8), qb4 = *(const v4f*)(qp + k*8 + 4);
    q8[0]=qa.x; q8[1]=qa.y; q8[2]=qa.z; q8[3]=qa.w;
    q8[4]=qb4.x; q8[5]=qb4.y; q8[6]=qb4.z; q8[7]=qb4.w;
    *(v4f*)(qp + k*8)     = (v4f){z[0], z[1], z[2], z[3]};    // z_k = pre-block state
    *(v4f*)(qp + k*8 + 4) = (v4f){z[4], z[5], z[6], z[7]};
#pragma unroll
    for (int i = 0; i < 8; ++i) {
      float acc = q8[i];
#pragma unroll
      for (int j = 0; j < 8; ++j) acc = fmaf(A[i*8 + j], z[j], acc);
      nz[i] = acc;
    }
#pragma unroll
    for (int i = 0; i < 8; ++i) z[i] = nz[i];
  }
}

// y_block = T*x_block + P*z_k  via 6 chained V_WMMA_F32_16X16X4_F32
template <int PASS>
__global__ void wmma_kernel(const float* __restrict__ ws, const float* __restrict__ x,
                            const float* __restrict__ y1v, const float* __restrict__ zbuf,
                            float* __restrict__ y1o, float* __restrict__ out, int band) {
  int wave = (blockIdx.x * blockDim.x + threadIdx.x) >> 5;
  if (wave >= 128 * KB) return;
  int cg = wave / KB, k = wave % KB;
  int lane = threadIdx.x & 31, half = lane >> 4, ln = lane & 15;
  int chain = cg * 16 + ln;      // WMMA N index = chain within group
  int t0 = k * 16;
  const float* cst = ws + band * CONST_F;

  // gather this lane's B-operand samples: slab s needs j = 4s+2*half, +1
  float xs[8];
  if (PASS == 0) {
    const float* row = x + (size_t)chain * TLEN;
    if (t0 >= PADL && t0 + 16 <= PADL + TLEN) {      // interior fast path (511/516 tiles)
      const float* p = row + (t0 - PADL) + 2 * half;
#pragma unroll
      for (int s = 0; s < 4; ++s) { xs[2*s] = p[4*s]; xs[2*s+1] = p[4*s + 1]; }
    } else {
#pragma unroll
      for (int s = 0; s < 4; ++s) {
        int j = t0 + 4*s + 2*half;
        xs[2*s] = load_ext(row, j); xs[2*s+1] = load_ext(row, j + 1);
      }
    }
  } else {
    const float* row = y1v + (size_t)chain * Y1STR;
    if (t0 + 16 <= TEXT) {                            // reversed fast path (515/516 tiles)
      const float* p = row + (TEXT - 1 - t0) - 2 * half;
#pragma unroll
      for (int s = 0; s < 4; ++s) { xs[2*s] = p[-4*s]; xs[2*s+1] = p[-4*s - 1]; }
    } else {
#pragma unroll
      for (int s = 0; s < 4; ++s) {
        int j = t0 + 4*s + 2*half;
        xs[2*s]   = (j     < TEXT) ? row[TEXT - 1 - j] : 0.0f;
        xs[2*s+1] = (j + 1 < TEXT) ? row[TEXT - 2 - j] : 0.0f;
      }
    }
  }

  v8f acc = {};
  // T (16x16) * X : 4 K-slabs of 4
#pragma unroll
  for (int s = 0; s < 4; ++s) {
    int k0 = 4 * s;
    v2f av, bv;
    const float* tp = cst + OFF_T + ln * 16 + k0 + 2 * half;  // A: M=ln, K pair
    av.x = tp[0]; av.y = tp[1];
    bv.x = xs[2*s]; bv.y = xs[2*s + 1];                       // B: K pair, N=ln
    acc = __builtin_amdgcn_wmma_f32_16x16x4_f32(false, av, false, bv,
                                                (short)0, acc, false, false);
  }
  // P (16x8) * z_k : 2 K-slabs of 4
#pragma unroll
  for (int s = 0; s < 2; ++s) {
    int k0 = 4 * s;
    v2f av, bv;
    const float* pp = cst + OFF_P + ln * 8 + k0 + 2 * half;
    av.x = pp[0]; av.y = pp[1];
    const float* zp = zbuf + (size_t)chain * QSTR + k * 8 + k0 + 2 * half;
    bv.x = zp[0]; bv.y = zp[1];
    acc = __builtin_amdgcn_wmma_f32_16x16x4_f32(false, av, false, bv,
                                                (short)0, acc, false, false);
  }

  // C/D layout: VGPR r -> M = r + 8*half, N = ln; each lane owns 8 consecutive t
  int tbase = t0 + 8 * half;
  if (PASS == 0) {
    float* wp = y1o + (size_t)chain * Y1STR + tbase;          // 32B-aligned
    if (tbase + 8 <= TEXT) {
      *(v4f*)(wp)     = (v4f){acc[0], acc[1], acc[2], acc[3]};
      *(v4f*)(wp + 4) = (v4f){acc[4], acc[5], acc[6], acc[7]};
    } else {
#pragma unroll
      for (int r = 0; r < 8; ++r) if (tbase + r < TEXT) wp[r] = acc[r];
    }
  } else {
    float* orow = out + (size_t)band * NCH * TLEN + (size_t)chain * TLEN;
    int obase = (TEXT - 1 - PADL) - tbase;                    // tout for r=0, descending
    if (obase - 7 >= 0 && obase < TLEN) {
      // obase-7 = 8211 - t0 - 8*half : 3 mod 4 floats -> not 16B-aligned
      float* wp = orow + (obase - 7);
      *(v4f*)(wp)     = (v4f){acc[7], acc[6], acc[5], acc[4]};
      *(v4f*)(wp + 4) = (v4f){acc[3], acc[2], acc[1], acc[0]};
    } else {
#pragma unroll
      for (int r = 0; r < 8; ++r) {
        int tout = obase - r;
        if (tout >= 0 && tout < TLEN) orow[tout] = acc[r];
      }
    }
  }
}

extern "C" void kernel_launch(void* const* d_in, const int* in_sizes, int n_in,
                              void* d_out, int out_size, void* d_ws, size_t ws_size,
                              hipStream_t stream) {
  const float* x    = (const float*)d_in[0];
  const float* sosl = (const float*)d_in[1];
  const float* sosh = (const float*)d_in[2];
  float* out = (float*)d_out;
  float* wsf = (float*)d_ws;

  float* y1 = wsf + DATA_OFF;                 // 2048 x 8256 f32  (~67.6 MB)
  float* qb = y1 + (size_t)NCH * Y1STR;       // 2048 x 516 x 8 f32 (~33.8 MB)
  (void)in_sizes; (void)n_in; (void)out_size; (void)ws_size;

  setup_kernel<<<1, 32, 0, stream>>>(wsf, sosl, sosh);

  dim3 qgrid((NCH * KB + 255) / 256);
  dim3 wgrid((128 * KB) / 8);                 // 8 waves / block, 1 tile / wave
  for (int band = 0; band < 2; ++band) {
    // pass 0: reflect-extended x -> y1
    q_kernel<0>   <<<qgrid, 256, 0, stream>>>(wsf, x, y1, qb, band);
    scan_kernel   <<<8,     256, 0, stream>>>(wsf, x, y1, qb, band, 0);
    wmma_kernel<0><<<wgrid, 256, 0, stream>>>(wsf, x, y1, qb, y1, out, band);
    // pass 1: reversed y1 -> reversed+trimmed directly into out
    q_kernel<1>   <<<qgrid, 256, 0, stream>>>(wsf, x, y1, qb, band);
    scan_kernel   <<<8,     256, 0, stream>>>(wsf, x, y1, qb, band, 1);
    wmma_kernel<1><<<wgrid, 256, 0, stream>>>(wsf, x, y1, qb, y1, out, band);
  }
}